// Scout_53377853555448
// MI455X (gfx1250) — compile-verified
//
#include <hip/hip_runtime.h>
#include <math.h>

// Problem constants (match reference)
#define DD 1024      // feature dim
#define BB 1024      // batch
#define KK 128       // retrieved rows per batch element

typedef __attribute__((ext_vector_type(2))) float v2f;
typedef __attribute__((ext_vector_type(8))) float v8f;

// ---------------------------------------------------------------------------
// GEMM: out[m][n] = act( sum_d A[m][d] * W[n][d] + bias[n] )
// CDNA5 V_WMMA_F32_16X16X4_F32 (fp32 tensor path, exact precision).
// Workgroup tile 64x128, 8 waves in 2x4 grid, each wave -> 32x32
// (four 16x16 accumulators, 2 A-frags x 2 B-frags => 1.0 LDS load per WMMA).
// ---------------------------------------------------------------------------
#define TILE_M 64
#define TILE_N 128
#define TILE_K 32
#define LDSTR  36   // padded LDS row stride (floats): 144B keeps float4 stores aligned, kills bank conflicts

__global__ __launch_bounds__(256)
void Scout_gemm_wmma_f32(const float* __restrict__ A,     // [1024][1024] row-major
                         const float* __restrict__ W,     // [1024][1024] row-major (used transposed)
                         const float* __restrict__ bias,  // [1024]
                         float* __restrict__ out,         // [1024][1024]
                         int applyGelu)
{
    __shared__ float sA[TILE_M * LDSTR];   //  9.2 KB
    __shared__ float sW[TILE_N * LDSTR];   // 18.4 KB

    const int t    = threadIdx.x;
    const int wave = t >> 5;        // 0..7
    const int lane = t & 31;
    const int wm   = wave >> 2;     // 0..1  (M sub-tile, 32 rows each)
    const int wn   = wave & 3;      // 0..3  (N sub-tile, 32 cols each)
    const int m0   = blockIdx.y * TILE_M;
    const int n0   = blockIdx.x * TILE_N;

    v8f c00 = {}, c01 = {}, c10 = {}, c11 = {};

    const int ma = 32 * wm + (lane & 15);   // local M row of first A fragment
    const int nb = 32 * wn + (lane & 15);   // local N row (of W) of first B fragment
    const int bk = 2 * (lane >> 4);         // K sub-offset per 16x16x4 layout

    for (int k0 = 0; k0 < DD; k0 += TILE_K) {
        // Cooperative tile load: A = 512 float4 (2/thread), W = 1024 float4 (4/thread).
#pragma unroll
        for (int i = 0; i < 2; ++i) {
            const int f   = t + i * 256;
            const int row = f >> 3;
            const int c4  = (f & 7) * 4;
            float4 va = *(const float4*)(A + (size_t)(m0 + row) * DD + k0 + c4);
            *(float4*)(&sA[row * LDSTR + c4]) = va;
        }
#pragma unroll
        for (int i = 0; i < 4; ++i) {
            const int f   = t + i * 256;
            const int row = f >> 3;
            const int c4  = (f & 7) * 4;
            float4 vw = *(const float4*)(W + (size_t)(n0 + row) * DD + k0 + c4);
            *(float4*)(&sW[row * LDSTR + c4]) = vw;
        }
        // Prefetch next K tile into cache while WMMAs drain (global_prefetch_b8)
        if (k0 + TILE_K < DD) {
            __builtin_prefetch(A + (size_t)(m0 + (t >> 3)) * DD + k0 + TILE_K + (t & 7) * 4, 0, 0);
            __builtin_prefetch(W + (size_t)(n0 + (t >> 1)) * DD + k0 + TILE_K + (t & 1) * 16, 0, 0);
        }
        __syncthreads();

#pragma unroll
        for (int kk = 0; kk < TILE_K; kk += 4) {
            // A fragments (16x4): lane l holds A[m][kk+bk], A[m][kk+bk+1]
            float2 av0 = *(const float2*)(&sA[ma * LDSTR + kk + bk]);
            float2 av1 = *(const float2*)(&sA[(ma + 16) * LDSTR + kk + bk]);
            // B fragments (4x16): lane l holds W[n][kk+bk], W[n][kk+bk+1]
            float2 bv0 = *(const float2*)(&sW[nb * LDSTR + kk + bk]);
            float2 bv1 = *(const float2*)(&sW[(nb + 16) * LDSTR + kk + bk]);
            v2f a0, a1, b0, b1;
            a0.x = av0.x; a0.y = av0.y;
            a1.x = av1.x; a1.y = av1.y;
            b0.x = bv0.x; b0.y = bv0.y;
            b1.x = bv1.x; b1.y = bv1.y;
            // 8-arg pattern: (neg_a, A, neg_b, B, c_mod, C, reuse_a, reuse_b)
            c00 = __builtin_amdgcn_wmma_f32_16x16x4_f32(false, a0, false, b0,
                                                        (short)0, c00, false, false);
            c01 = __builtin_amdgcn_wmma_f32_16x16x4_f32(false, a0, false, b1,
                                                        (short)0, c01, false, false);
            c10 = __builtin_amdgcn_wmma_f32_16x16x4_f32(false, a1, false, b0,
                                                        (short)0, c10, false, false);
            c11 = __builtin_amdgcn_wmma_f32_16x16x4_f32(false, a1, false, b1,
                                                        (short)0, c11, false, false);
        }
        __syncthreads();
    }

    // Writeback. C/D layout: VGPR r -> M = r (lanes 0-15) / 8+r (lanes 16-31), N = lane%16.
    const int mbase = m0 + 32 * wm + 8 * (lane >> 4);
    const int ncol  = n0 + 32 * wn + (lane & 15);
    const float bia0 = bias[ncol];
    const float bia1 = bias[ncol + 16];
#pragma unroll
    for (int r = 0; r < 8; ++r) {
        const int mA = mbase + r;        // rows of c00/c01
        const int mB = mbase + 16 + r;   // rows of c10/c11
        float v00 = c00[r] + bia0;
        float v01 = c01[r] + bia1;
        float v10 = c10[r] + bia0;
        float v11 = c11[r] + bia1;
        if (applyGelu) {
            v00 = 0.5f * v00 * (1.0f + erff(v00 * 0.70710678118654752f));
            v01 = 0.5f * v01 * (1.0f + erff(v01 * 0.70710678118654752f));
            v10 = 0.5f * v10 * (1.0f + erff(v10 * 0.70710678118654752f));
            v11 = 0.5f * v11 * (1.0f + erff(v11 * 0.70710678118654752f));
        }
        out[(size_t)mA * DD + ncol]      = v00;
        out[(size_t)mA * DD + ncol + 16] = v01;
        out[(size_t)mB * DD + ncol]      = v10;
        out[(size_t)mB * DD + ncol + 16] = v11;
    }
}

// ---------------------------------------------------------------------------
// Fused gather + scores + softmax + recombine. One workgroup per batch row b.
// Phase 1: wave-per-row dot(memory[idx], q[b]) with q in LDS (streams 512KB from HBM)
// Phase 2: softmax over 128 scores in LDS, write P0
// Phase 3: per-thread float4 of Y0[b], accumulate over 128 rows (L2-hot re-read)
// ---------------------------------------------------------------------------
__global__ __launch_bounds__(256)
void Scout_attn_fused(const float* __restrict__ memory,     // [N_MEM][1024]
                      const long long* __restrict__ Kset,   // [B][K] int64
                      const float* __restrict__ queries,    // [B][1024]
                      float* __restrict__ P0,               // [B][K]
                      float* __restrict__ Y0)               // [B][1024]
{
    __shared__ float qs[DD];       // 4 KB: query row
    __shared__ float sc[KK];       // scores -> probs
    __shared__ int   sidx[KK];     // gathered row indices
    __shared__ float red[64];      // reduction scratch

    const int b    = blockIdx.x;
    const int t    = threadIdx.x;   // 0..255
    const int wave = t >> 5;        // 0..7
    const int lane = t & 31;

    // Load q[b] and indices into LDS
    ((float4*)qs)[t] = ((const float4*)(queries + (size_t)b * DD))[t];
    if (t < KK) sidx[t] = (int)Kset[(size_t)b * KK + t];
    __syncthreads();

    // ---- Phase 1: scores[k] = dot(memory[sidx[k]], q) ----
#pragma unroll
    for (int it = 0; it < KK / 8; ++it) {
        const int k = it * 8 + wave;
        const float4* row = (const float4*)(memory + (size_t)sidx[k] * DD);
        float s = 0.0f;
#pragma unroll
        for (int i = 0; i < 8; ++i) {
            const int j = i * 32 + lane;      // 256 float4 per row, 8 per lane
            float4 r  = row[j];
            float4 qv = ((const float4*)qs)[j];
            s += r.x * qv.x + r.y * qv.y + r.z * qv.z + r.w * qv.w;
        }
#pragma unroll
        for (int off = 16; off; off >>= 1) s += __shfl_xor(s, off, 32);
        if (lane == 0) sc[k] = s;
    }
    __syncthreads();

    // ---- Phase 2: softmax over sc[0..127] ----
    if (t < 64) red[t] = fmaxf(sc[t], sc[t + 64]);
    __syncthreads();
    if (t < 32) {
        float v = fmaxf(red[t], red[t + 32]);
#pragma unroll
        for (int off = 16; off; off >>= 1) v = fmaxf(v, __shfl_xor(v, off, 32));
        if (t == 0) red[0] = v;
    }
    __syncthreads();
    const float smax = red[0];
    if (t < KK) sc[t] = __expf(sc[t] - smax);
    __syncthreads();
    if (t < 64) red[t] = sc[t] + sc[t + 64];
    __syncthreads();
    if (t < 32) {
        float v = red[t] + red[t + 32];
#pragma unroll
        for (int off = 16; off; off >>= 1) v += __shfl_xor(v, off, 32);
        if (t == 0) red[0] = v;
    }
    __syncthreads();
    const float inv = 1.0f / red[0];
    if (t < KK) {
        const float p = sc[t] * inv;
        sc[t] = p;
        P0[(size_t)b * KK + t] = p;
    }
    __syncthreads();

    // ---- Phase 3: Y0[b] = sum_k p[k] * memory[sidx[k]]  (rows L2-hot) ----
    float4 acc = {0.0f, 0.0f, 0.0f, 0.0f};
#pragma unroll 4
    for (int k = 0; k < KK; ++k) {
        const float4* row = (const float4*)(memory + (size_t)sidx[k] * DD);
        const float4 r = row[t];
        const float  p = sc[k];
        acc.x += p * r.x;
        acc.y += p * r.y;
        acc.z += p * r.z;
        acc.w += p * r.w;
    }
    ((float4*)(Y0 + (size_t)b * DD))[t] = acc;
}

// ---------------------------------------------------------------------------
extern "C" void kernel_launch(void* const* d_in, const int* in_sizes, int n_in,
                              void* d_out, int out_size, void* d_ws, size_t ws_size,
                              hipStream_t stream) {
    const float*     emb  = (const float*)d_in[0];      // [B][D]
    const float*     mem  = (const float*)d_in[1];      // [N_MEM][D]
    const long long* kset = (const long long*)d_in[2];  // [B][K] int64
    const float*     W1   = (const float*)d_in[3];      // [D][D]
    const float*     b1   = (const float*)d_in[4];      // [D]
    const float*     W2   = (const float*)d_in[5];      // [D][D]
    const float*     b2   = (const float*)d_in[6];      // [D]

    float* P0 = (float*)d_out;                          // [B][K]
    float* Y0 = P0 + (size_t)BB * KK;                   // [B][D]

    float* H = (float*)d_ws;                            // [B][D] hidden
    float* Q = H + (size_t)BB * DD;                     // [B][D] queries

    dim3 gg(DD / TILE_N, BB / TILE_M);                  // 8 x 16 blocks
    Scout_gemm_wmma_f32<<<gg, 256, 0, stream>>>(emb, W1, b1, H, 1);
    Scout_gemm_wmma_f32<<<gg, 256, 0, stream>>>(H, W2, b2, Q, 0);
    Scout_attn_fused<<<BB, 256, 0, stream>>>(mem, kset, Q, P0, Y0);
}